// HybridGATLSTM_8693013807251
// MI455X (gfx1250) — compile-verified
//
#include <hip/hip_runtime.h>
#include <hip/hip_bf16.h>
#include <math.h>

#define B_   8
#define S_   24
#define N_   400
#define E_   6400
#define H_   128
#define EN_  (E_ + N_)      // 6800 edges incl. self loops
#define G4H_ 512            // 4*H

typedef float v2f __attribute__((ext_vector_type(2)));
typedef float v8f __attribute__((ext_vector_type(8)));

// ---------------- order-preserving float<->uint for atomicMax ----------------
__device__ __forceinline__ unsigned fenc(float f) {
    unsigned b = __float_as_uint(f);
    return (b & 0x80000000u) ? ~b : (b | 0x80000000u);
}
__device__ __forceinline__ float fdec(unsigned u) {
    unsigned b = (u & 0x80000000u) ? (u ^ 0x80000000u) : ~u;
    return __uint_as_float(b);
}
__device__ __forceinline__ float sigmoidf_(float x) { return 1.0f / (1.0f + expf(-x)); }

// ------------------------------------------------------------------
// Wave-level GEMM accumulate: acc += A(16xK=128, LDS) * W^T tile.
//   Alds : 16 rows x 128, row stride 128 (rows 8..15 zero-padded)
//   Wg   : global, row-major (rows of length 128); B[kk][n] = Wg[(n0+n)*128 + k0+kk]
// Uses V_WMMA_F32_16X16X4_F32, 32 k-steps. Must be called wave-uniformly.
// ------------------------------------------------------------------
__device__ __forceinline__ v8f wave_gemm_k128(const float* __restrict__ Alds,
                                              const float* __restrict__ Wg,
                                              int n0, int lane, v8f acc) {
    const int half = (lane >> 4) & 1;      // 0: K pair {0,1}, 1: K pair {2,3}
    const int mr   = lane & 15;            // A row (M) / B col (N)
    #pragma unroll 4
    for (int k0 = 0; k0 < H_; k0 += 4) {
        const int ka = k0 + (half ? 2 : 0);
        v2f a, b;
        a.x = Alds[mr * H_ + ka];
        a.y = Alds[mr * H_ + ka + 1];
        const float* wrow = Wg + (size_t)(n0 + mr) * H_ + ka;
        b.x = wrow[0];
        b.y = wrow[1];
        acc = __builtin_amdgcn_wmma_f32_16x16x4_f32(
            /*neg_a=*/false, a, /*neg_b=*/false, b,
            /*c_mod=*/(short)0, acc, /*reuse_a=*/false, /*reuse_b=*/false);
    }
    return acc;
}

// ==================================================================
// Stage 1: GAT per (b,t) graph. 192 blocks, 256 threads.
// emb layout: (S, B, H)
// ==================================================================
__global__ __launch_bounds__(256)
void gat_kernel(const float* __restrict__ x, const int* __restrict__ ei,
                const float* __restrict__ gat_w, const float* __restrict__ att_src,
                const float* __restrict__ att_dst, const float* __restrict__ gat_bias,
                float* __restrict__ emb) {
    __shared__ float    xs[N_];
    __shared__ unsigned mx[N_];
    __shared__ float    ssum[N_];
    __shared__ float    sacc[N_];
    __shared__ float    elog[EN_];

    const int tid = threadIdx.x;
    const int bt  = blockIdx.x;
    const int b   = bt / S_;
    const int t   = bt % S_;

    // rank-1 collapse: a_s = cs*x, a_d = cd*x
    float cs = 0.f, cd = 0.f;
    for (int k = 0; k < H_; ++k) {
        const float g = gat_w[k];
        cs += g * att_src[k];
        cd += g * att_dst[k];
    }

    const float* xrow = x + ((size_t)b * S_ + t) * N_;
    for (int n = tid; n < N_; n += blockDim.x) {
        xs[n]   = xrow[n];
        mx[n]   = 0u;           // decodes below any real float
        ssum[n] = 0.f;
        sacc[n] = 0.f;
    }
    __syncthreads();

    // phase 1: segment max of leaky-relu logits
    for (int e = tid; e < EN_; e += blockDim.x) {
        const int s = (e < E_) ? ei[e]       : (e - E_);
        const int d = (e < E_) ? ei[E_ + e]  : (e - E_);
        float lg = cs * xs[s] + cd * xs[d];
        lg = (lg >= 0.f) ? lg : 0.2f * lg;
        atomicMax(&mx[d], fenc(lg));
    }
    __syncthreads();

    // phase 2: exp(logit - max) and segment sum
    for (int e = tid; e < EN_; e += blockDim.x) {
        const int s = (e < E_) ? ei[e]       : (e - E_);
        const int d = (e < E_) ? ei[E_ + e]  : (e - E_);
        float lg = cs * xs[s] + cd * xs[d];
        lg = (lg >= 0.f) ? lg : 0.2f * lg;
        const float ex = expf(lg - fdec(mx[d]));
        elog[e] = ex;
        atomicAdd(&ssum[d], ex);
    }
    __syncthreads();

    // phase 3: s[n] = sum alpha * x[src]
    for (int e = tid; e < EN_; e += blockDim.x) {
        const int s = (e < E_) ? ei[e]       : (e - E_);
        const int d = (e < E_) ? ei[E_ + e]  : (e - E_);
        const float alpha = elog[e] / ssum[d];
        atomicAdd(&sacc[d], alpha * xs[s]);
    }
    __syncthreads();

    // phase 4: emb[h] = mean_n relu(s[n]*gw[h] + gb[h])
    for (int h = tid; h < H_; h += blockDim.x) {
        const float gw = gat_w[h];
        const float gb = gat_bias[h];
        float acc = 0.f;
        for (int n = 0; n < N_; ++n) {
            const float v = sacc[n] * gw + gb;
            acc += (v > 0.f) ? v : 0.f;
        }
        emb[((size_t)t * B_ + b) * H_ + h] = acc * (1.0f / (float)N_);
    }
}

// ==================================================================
// Stage 2: 2-layer LSTM, sequential. One block, 256 threads (8 waves).
// Gate GEMMs via V_WMMA_F32_16X16X4_F32 (M padded 8->16).
// ==================================================================
__global__ __launch_bounds__(256)
void lstm_kernel(const float* __restrict__ emb,
                 const float* __restrict__ wih0, const float* __restrict__ whh0,
                 const float* __restrict__ bih0, const float* __restrict__ bhh0,
                 const float* __restrict__ wih1, const float* __restrict__ whh1,
                 const float* __restrict__ bih1, const float* __restrict__ bhh1,
                 float* __restrict__ hs0, float* __restrict__ zbuf) {
    __shared__ float X [16 * H_];
    __shared__ float Hp[16 * H_];
    __shared__ float Cc[16 * H_];
    __shared__ float G [16 * G4H_];

    const int tid  = threadIdx.x;
    const int lane = tid & 31;
    const int wave = tid >> 5;

    for (int L = 0; L < 2; ++L) {
        const float* wih   = L ? wih1 : wih0;
        const float* whh   = L ? whh1 : whh0;
        const float* bih   = L ? bih1 : bih0;
        const float* bhh   = L ? bhh1 : bhh0;
        const float* inbuf = L ? hs0  : emb;

        for (int i = tid; i < 16 * H_; i += blockDim.x) { Hp[i] = 0.f; Cc[i] = 0.f; }
        __syncthreads();

        for (int t = 0; t < S_; ++t) {
            // load x_t (rows 8..15 zero)
            for (int i = tid; i < 16 * H_; i += blockDim.x) {
                const int m = i / H_;
                X[i] = (m < B_) ? inbuf[(size_t)t * B_ * H_ + i] : 0.f;
            }
            __syncthreads();

            // G = X @ wih^T + Hp @ whh^T ; 32 n-tiles, 4 per wave (uniform)
            for (int nt = wave * 4; nt < wave * 4 + 4; ++nt) {
                const int n0 = nt * 16;
                v8f acc = {};
                acc = wave_gemm_k128(X,  wih, n0, lane, acc);
                acc = wave_gemm_k128(Hp, whh, n0, lane, acc);
                const int mbase = (lane >= 16) ? 8 : 0;
                const int nn    = n0 + (lane & 15);
                #pragma unroll
                for (int v = 0; v < 8; ++v) G[(mbase + v) * G4H_ + nn] = acc[v];
            }
            __syncthreads();

            // elementwise gates (i,f,g,o)
            for (int idx = tid; idx < B_ * H_; idx += blockDim.x) {
                const int m = idx / H_;
                const int h = idx % H_;
                const float gi = G[m * G4H_ + h]            + bih[h]           + bhh[h];
                const float gf = G[m * G4H_ + H_ + h]       + bih[H_ + h]      + bhh[H_ + h];
                const float gg = G[m * G4H_ + 2 * H_ + h]   + bih[2 * H_ + h]  + bhh[2 * H_ + h];
                const float go = G[m * G4H_ + 3 * H_ + h]   + bih[3 * H_ + h]  + bhh[3 * H_ + h];
                const float c  = sigmoidf_(gf) * Cc[m * H_ + h] + sigmoidf_(gi) * tanhf(gg);
                const float hh = sigmoidf_(go) * tanhf(c);
                Cc[m * H_ + h] = c;
                Hp[m * H_ + h] = hh;
                if (L == 0)               hs0 [(size_t)t * B_ * H_ + idx] = hh;
                else if (t == S_ - 1)     zbuf[idx] = hh;
            }
            __syncthreads();
        }
    }
}

// ==================================================================
// Stage 3: heads p = z@price_w^T + b, r = z@return_w^T + b.
// N=400 = 25 WMMA n-tiles per head; 50 tiles over 8 waves.
// ==================================================================
__global__ __launch_bounds__(256)
void head_kernel(const float* __restrict__ zbuf,
                 const float* __restrict__ pw, const float* __restrict__ pb,
                 const float* __restrict__ rw, const float* __restrict__ rb,
                 float* __restrict__ out) {
    __shared__ float Z[16 * H_];
    const int tid  = threadIdx.x;
    const int lane = tid & 31;
    const int wave = tid >> 5;

    for (int i = tid; i < 16 * H_; i += blockDim.x) {
        const int m = i / H_;
        Z[i] = (m < B_) ? zbuf[i] : 0.f;
    }
    __syncthreads();

    for (int tile = wave; tile < 50; tile += 8) {   // uniform per wave
        const int head = tile / 25;
        const int n0   = (tile % 25) * 16;
        const float* W    = head ? rw : pw;
        const float* bias = head ? rb : pb;
        v8f acc = {};
        acc = wave_gemm_k128(Z, W, n0, lane, acc);
        if (lane < 16) {                             // rows 0..7 only (M=8)
            const int n  = n0 + lane;
            const float bv = bias[n];
            #pragma unroll
            for (int v = 0; v < 8; ++v)
                out[(size_t)head * B_ * N_ + (size_t)v * N_ + n] = acc[v] + bv;
        }
    }
}

extern "C" void kernel_launch(void* const* d_in, const int* in_sizes, int n_in,
                              void* d_out, int out_size, void* d_ws, size_t ws_size,
                              hipStream_t stream) {
    const float* x        = (const float*)d_in[0];
    const int*   ei       = (const int*)  d_in[1];
    const float* gat_w    = (const float*)d_in[2];
    const float* att_src  = (const float*)d_in[3];
    const float* att_dst  = (const float*)d_in[4];
    const float* gat_bias = (const float*)d_in[5];
    const float* wih0     = (const float*)d_in[6];
    const float* whh0     = (const float*)d_in[7];
    const float* bih0     = (const float*)d_in[8];
    const float* bhh0     = (const float*)d_in[9];
    const float* wih1     = (const float*)d_in[10];
    const float* whh1     = (const float*)d_in[11];
    const float* bih1     = (const float*)d_in[12];
    const float* bhh1     = (const float*)d_in[13];
    const float* pw       = (const float*)d_in[14];
    const float* pb       = (const float*)d_in[15];
    const float* rw       = (const float*)d_in[16];
    const float* rb       = (const float*)d_in[17];
    float* out = (float*)d_out;

    float* ws   = (float*)d_ws;
    float* emb  = ws;                       // S*B*H = 24576
    float* hs0  = ws + (size_t)S_ * B_ * H_;   // + 24576
    float* zbuf = hs0 + (size_t)S_ * B_ * H_;  // + 24576 (uses B*H)

    gat_kernel<<<B_ * S_, 256, 0, stream>>>(x, ei, gat_w, att_src, att_dst, gat_bias, emb);
    lstm_kernel<<<1, 256, 0, stream>>>(emb, wih0, whh0, bih0, bhh0,
                                       wih1, whh1, bih1, bhh1, hs0, zbuf);
    head_kernel<<<1, 256, 0, stream>>>(zbuf, pw, pb, rw, rb, out);
}